// Head_61787399520261
// MI455X (gfx1250) — compile-verified
//
#include <hip/hip_runtime.h>

typedef __attribute__((ext_vector_type(2))) float v2f;
typedef __attribute__((ext_vector_type(4))) float v4f;
typedef __attribute__((ext_vector_type(8))) float v8f;
typedef __attribute__((ext_vector_type(4))) unsigned v4u;
typedef __attribute__((ext_vector_type(4))) int v4i;
typedef __attribute__((ext_vector_type(8))) int v8i;

#define WMMA_F32(a, b, c) \
  __builtin_amdgcn_wmma_f32_16x16x4_f32(false, (a), false, (b), (short)0, (c), false, false)

static constexpr int B_ = 128, T_ = 256, C_ = 1024, H_ = 64;
static constexpr int BTH = B_ * T_ * H_;

// -----------------------------------------------------------------------------
// TDM helper: async-load a [tile_d1=16 rows x 64 dwords] f32 tile to LDS with
// 4-dword row padding (LDS row stride = 68 floats). D# per CDNA5 ISA:
//   group0: count=1 | lds_addr | global_addr(57b) | type=2
//   group1: data_size=4B, pad_enable, pad_interval=5 (64 dw), pad_amount=3 (4 dw),
//           tensor_dim0/1, tile_dim0=64, tile_dim1=16, tensor_dim0_stride
//   groups 2/3: zero (2D tensor)
// -----------------------------------------------------------------------------
__device__ __forceinline__ void tdm_load_tile64(const float* g, unsigned lds_byte,
                                                unsigned dim0, unsigned dim1,
                                                unsigned stride0) {
  unsigned long long ga = (unsigned long long)(uintptr_t)g;
  v4u g0 = {1u,                                   // count=1 (valid descriptor)
            lds_byte,                             // lds_addr
            (unsigned)(ga & 0xFFFFFFFFu),         // global_addr[31:0]
            (unsigned)((ga >> 32) & 0x1FFFFFFu) | (2u << 30)};  // addr[56:32]|type=2
  v8i g1 = {(int)((2u << 16) | (1u << 20) | (5u << 22) | (3u << 25)),
            (int)((dim0 & 0xFFFFu) << 16),
            (int)(((dim0 >> 16) & 0xFFFFu) | ((dim1 & 0xFFFFu) << 16)),
            (int)(((dim1 >> 16) & 0xFFFFu) | (64u << 16)),   // tile_dim0 = 64
            (int)16,                                          // tile_dim1 = 16
            (int)stride0,                                     // tensor_dim0_stride lo
            0, 0};
  v4i gz = {0, 0, 0, 0};
#if defined(__clang_major__) && (__clang_major__ >= 23)
  v8i gz8 = {0, 0, 0, 0, 0, 0, 0, 0};
  __builtin_amdgcn_tensor_load_to_lds(g0, g1, gz, gz, gz8, 0);
#else
  __builtin_amdgcn_tensor_load_to_lds(g0, g1, gz, gz, 0);
#endif
}

// -----------------------------------------------------------------------------
// Kernel 1: fused QKV projection.  [B*T, C] x [C, 3*H] -> q,k,v [B*T, H]
// 512 blocks x 128 threads (4 waves). Each block: 64 rows, all 192 cols.
// x tiles arrive via double-buffered TDM; W chunks staged transposed manually.
// -----------------------------------------------------------------------------
__global__ __launch_bounds__(128) void qkv_proj_kernel(
    const float* __restrict__ x, const float* __restrict__ Wq,
    const float* __restrict__ Wk, const float* __restrict__ Wv,
    float* __restrict__ qb, float* __restrict__ kbuf, float* __restrict__ vbuf)
{
  __shared__ float xs[2 * 4 * 16 * 68];  // double-buffered per-wave 16x64 tile, pad 68
  __shared__ float wt[3 * 64 * 66];      // W chunk transposed: [mat][n=64][k=64], pad 66

  const int tid  = threadIdx.x;
  const int w    = tid >> 5;
  const int lane = tid & 31;
  const int n16  = lane & 15;
  const int off  = (lane < 16) ? 0 : 2;
  const int wrow0 = blockIdx.x * 64 + w * 16;

  v8f acc[3][4];
  #pragma unroll
  for (int m = 0; m < 3; ++m)
    #pragma unroll
    for (int t = 0; t < 4; ++t)
      acc[m][t] = (v8f){0.f, 0.f, 0.f, 0.f, 0.f, 0.f, 0.f, 0.f};

  // Kick off the DMA for the first x K-block into buffer 0.
  tdm_load_tile64(&x[(size_t)wrow0 * C_],
                  (unsigned)(uintptr_t)&xs[(0 * 4 + w) * 16 * 68],
                  (unsigned)C_, (unsigned)(B_ * T_), (unsigned)C_);

  int cur = 0;
  #pragma unroll 1
  for (int kb0 = 0; kb0 < C_; kb0 += 64) {
    // Prefetch next x K-block into the other buffer while we work.
    if (kb0 + 64 < C_)
      tdm_load_tile64(&x[(size_t)wrow0 * C_ + kb0 + 64],
                      (unsigned)(uintptr_t)&xs[((cur ^ 1) * 4 + w) * 16 * 68],
                      (unsigned)C_, (unsigned)(B_ * T_), (unsigned)C_);

    // Stage W chunk transposed: 3 mats x 64k x 64n = 3072 vec4 over 128 threads.
    // Overlaps with the in-flight tensor DMA.
    #pragma unroll
    for (int i = 0; i < 24; ++i) {
      int f   = tid + i * 128;
      int mat = f >> 10;
      int rem = f & 1023;
      int kr  = rem >> 4;
      int c4  = (rem & 15) << 2;
      const float* wp = (mat == 0) ? Wq : ((mat == 1) ? Wk : Wv);
      v4f wv = *(const v4f*)&wp[(size_t)(kb0 + kr) * H_ + c4];
      wt[(mat * 64 + c4 + 0) * 66 + kr] = wv.x;
      wt[(mat * 64 + c4 + 1) * 66 + kr] = wv.y;
      wt[(mat * 64 + c4 + 2) * 66 + kr] = wv.z;
      wt[(mat * 64 + c4 + 3) * 66 + kr] = wv.w;
      if (kb0 + 64 < C_)
        __builtin_prefetch(&wp[(size_t)(kb0 + 64 + kr) * H_ + c4], 0, 2);
    }

    // Wait for the *current* buffer's DMA (the prefetch may stay in flight).
    if (kb0 + 64 < C_) __builtin_amdgcn_s_wait_tensorcnt(1);
    else               __builtin_amdgcn_s_wait_tensorcnt(0);
    __syncthreads();

    // 16 K-chunks of 4 -> 192 WMMAs per K block per wave.
    #pragma unroll
    for (int kc = 0; kc < 64; kc += 4) {
      v2f a = *(const v2f*)&xs[((cur * 4 + w) * 16 + n16) * 68 + kc + off];
      #pragma unroll
      for (int mat = 0; mat < 3; ++mat)
        #pragma unroll
        for (int t = 0; t < 4; ++t) {
          v2f bb = *(const v2f*)&wt[(mat * 64 + t * 16 + n16) * 66 + kc + off];
          acc[mat][t] = WMMA_F32(a, bb, acc[mat][t]);
        }
    }
    __syncthreads();
    cur ^= 1;
  }

  float* outs[3] = {qb, kbuf, vbuf};
  #pragma unroll
  for (int mat = 0; mat < 3; ++mat) {
    float* op = outs[mat];
    #pragma unroll
    for (int t = 0; t < 4; ++t)
      #pragma unroll
      for (int r = 0; r < 8; ++r) {
        int gr = wrow0 + ((lane < 16) ? r : r + 8);
        op[(size_t)gr * H_ + t * 16 + n16] = acc[mat][t][r];
      }
  }
}

// -----------------------------------------------------------------------------
// Kernel 2: flash attention. One wave per (batch, 16-query tile).
// k block arrives via TDM (overlapped with manual v-transpose staging).
// -----------------------------------------------------------------------------
__global__ __launch_bounds__(32) void attn_kernel(
    const float* __restrict__ qb, const float* __restrict__ kbuf,
    const float* __restrict__ vbuf, float* __restrict__ out)
{
  __shared__ float ks[16 * 68];  // k block, row-major, pad 68 (TDM pad config)
  __shared__ float vt[64 * 18];  // v block transposed: [h=64][s=16], pad 18
  __shared__ float ps[16 * 18];  // P tile for C-layout -> A-layout round trip

  const int lane = threadIdx.x;
  const int n    = lane & 15;
  const int off  = (lane < 16) ? 0 : 2;
  const int b    = blockIdx.x >> 4;
  const int it   = blockIdx.x & 15;
  const int t0   = it * 16;
  const float scale = 0.125f;  // 1/sqrt(64)

  // q tile resident in A-layout registers, pre-scaled.
  v2f aq[16];
  #pragma unroll
  for (int c = 0; c < 16; ++c) {
    v2f qv = *(const v2f*)&qb[(size_t)(b * T_ + t0 + n) * H_ + 4 * c + off];
    aq[c] = qv * scale;
  }

  v8f acco[4];
  #pragma unroll
  for (int t = 0; t < 4; ++t)
    acco[t] = (v8f){0.f, 0.f, 0.f, 0.f, 0.f, 0.f, 0.f, 0.f};
  float mrow[8], lrow[8];
  #pragma unroll
  for (int r = 0; r < 8; ++r) { mrow[r] = -__builtin_inff(); lrow[r] = 0.f; }

  const int rbase = t0 + ((lane < 16) ? 0 : 8);
  const unsigned ks_lds = (unsigned)(uintptr_t)&ks[0];

  #pragma unroll 1
  for (int jb = 0; jb <= it; ++jb) {
    const int s0 = jb * 16;

    // Async TDM load of the k block (16 x 64, stride H, pad -> 68-float rows).
    tdm_load_tile64(&kbuf[(size_t)(b * T_ + s0) * H_], ks_lds,
                    (unsigned)H_, (unsigned)(B_ * T_), (unsigned)H_);

    // Meanwhile stage v transposed (coalesced b128 loads, scattered b32 stores).
    #pragma unroll
    for (int i = 0; i < 8; ++i) {
      int f  = lane + i * 32;
      int r  = f >> 4;
      int c4 = (f & 15) << 2;
      v4f vv = *(const v4f*)&vbuf[(size_t)(b * T_ + s0 + r) * H_ + c4];
      vt[(c4 + 0) * 18 + r] = vv.x;
      vt[(c4 + 1) * 18 + r] = vv.y;
      vt[(c4 + 2) * 18 + r] = vv.z;
      vt[(c4 + 3) * 18 + r] = vv.w;
    }
    __builtin_amdgcn_s_wait_tensorcnt(0);
    __syncthreads();

    // S = (q*scale) k^T : 16 WMMAs over H=64.
    v8f s = (v8f){0.f, 0.f, 0.f, 0.f, 0.f, 0.f, 0.f, 0.f};
    #pragma unroll
    for (int c = 0; c < 16; ++c) {
      v2f bk = *(const v2f*)&ks[n * 68 + 4 * c + off];
      s = WMMA_F32(aq[c], bk, s);
    }

    // Causal mask + online softmax (segment reductions over the 16-lane halves).
    #pragma unroll
    for (int r = 0; r < 8; ++r) {
      float sv = s[r];
      sv = (s0 + n <= rbase + r) ? sv : -__builtin_inff();
      float rmax = sv;
      #pragma unroll
      for (int o = 8; o >= 1; o >>= 1) rmax = fmaxf(rmax, __shfl_xor(rmax, o, 16));
      float nm = fmaxf(mrow[r], rmax);
      float p  = __expf(sv - nm);
      float rs = p;
      #pragma unroll
      for (int o = 8; o >= 1; o >>= 1) rs += __shfl_xor(rs, o, 16);
      float corr = __expf(mrow[r] - nm);
      lrow[r] = lrow[r] * corr + rs;
      mrow[r] = nm;
      ps[((lane < 16) ? r : r + 8) * 18 + n] = p;
      #pragma unroll
      for (int t = 0; t < 4; ++t) acco[t][r] *= corr;
    }
    __syncthreads();

    // out += P @ v : 64 WMMAs (P read back from LDS in A layout).
    #pragma unroll
    for (int t = 0; t < 4; ++t) {
      int h = t * 16 + n;
      v8f a8 = acco[t];
      #pragma unroll
      for (int c = 0; c < 16; ++c) {
        v2f ap = *(const v2f*)&ps[n * 18 + 4 * c + off];
        v2f bv = *(const v2f*)&vt[h * 18 + 4 * c + off];
        a8 = WMMA_F32(ap, bv, a8);
      }
      acco[t] = a8;
    }
    __syncthreads();
  }

  // Normalize and store.
  #pragma unroll
  for (int t = 0; t < 4; ++t)
    #pragma unroll
    for (int r = 0; r < 8; ++r) {
      int gr = t0 + ((lane < 16) ? r : r + 8);
      out[(size_t)(b * T_ + gr) * H_ + t * 16 + n] = acco[t][r] / lrow[r];
    }
}

extern "C" void kernel_launch(void* const* d_in, const int* in_sizes, int n_in,
                              void* d_out, int out_size, void* d_ws, size_t ws_size,
                              hipStream_t stream) {
  (void)in_sizes; (void)n_in; (void)out_size; (void)ws_size;
  const float* x  = (const float*)d_in[0];
  const float* Wk = (const float*)d_in[1];
  const float* Wq = (const float*)d_in[2];
  const float* Wv = (const float*)d_in[3];
  float* ws = (float*)d_ws;
  float* qb = ws;
  float* kb = ws + BTH;
  float* vb = ws + 2 * (size_t)BTH;

  qkv_proj_kernel<<<dim3((B_ * T_) / 64), dim3(128), 0, stream>>>(
      x, Wq, Wk, Wv, qb, kb, vb);
  attn_kernel<<<dim3(B_ * (T_ / 16)), dim3(32), 0, stream>>>(
      qb, kb, vb, (float*)d_out);
}